// VolumeRenderer_49426483642806
// MI455X (gfx1250) — compile-verified
//
#include <hip/hip_runtime.h>

// Volume rendering (NeRF composite): R rays x S samples, sequential scan per ray.
// Memory-bound: ~252 MB streamed -> ~11 us floor at 23.3 TB/s on MI455X.
// Strategy: one wave32 per ray; wave-level shuffle scan for the cumsum;
// density stream staged into LDS via CDNA5 async-load-to-LDS (ASYNCcnt),
// all 6 chunks issued up front for max MLP; color stream prefetched to WGP$
// and loaded as b96; no WMMA (no matmul structure in this op).

#define R_RAYS    65536
#define S_SAMP    192
#define N_CHUNKS  6          // 192 / 32
#define FARDELTA  1e10f
#define WPB       8          // waves per block (256 threads)

// Low 32 bits of a generic pointer into LDS are the raw LDS byte offset
// (ISA: aperture check uses addr[63:32]; the low half is the LDS address).
__device__ __forceinline__ unsigned lds_off_u32(const void* p) {
    return (unsigned)(unsigned long long)p;
}

__device__ __forceinline__ void async_load_f32_to_lds(const float* gaddr, unsigned lds_off) {
    // GLOBAL_LOAD_ASYNC_TO_LDS_B32, GV mode (saddr = off). Per-lane:
    // copies 4 bytes from the global address in gaddr to the LDS offset in lds_off. Tracked with ASYNCcnt.
    asm volatile("global_load_async_to_lds_b32 %0, %1, off"
                 :: "v"(lds_off), "v"(gaddr)
                 : "memory");
}

// s_wait_asynccnt takes an immediate only; after full unroll the switch folds
// to a single instruction.
__device__ __forceinline__ void wait_async_le(int n) {
    switch (n) {
    case 0: asm volatile("s_wait_asynccnt 0" ::: "memory"); break;
    case 1: asm volatile("s_wait_asynccnt 1" ::: "memory"); break;
    case 2: asm volatile("s_wait_asynccnt 2" ::: "memory"); break;
    case 3: asm volatile("s_wait_asynccnt 3" ::: "memory"); break;
    case 4: asm volatile("s_wait_asynccnt 4" ::: "memory"); break;
    default: asm volatile("s_wait_asynccnt 5" ::: "memory"); break;
    }
}

// 12-byte, 4-aligned RGB triple: lets the backend emit global_load_b96.
struct __attribute__((aligned(4))) F3 { float x, y, z; };

__global__ __launch_bounds__(256) void vr_pass1(
    const float* __restrict__ density,   // [R, S]
    const float* __restrict__ color,     // [R, S, 3]
    const float* __restrict__ depth,     // [R, S]
    float*       __restrict__ out,       // [R, 4]
    unsigned*    __restrict__ ws_max)    // global max(dep) as ordered u32 bits
{
    __shared__ float sden[N_CHUNKS][WPB][32];  // full density stream per wave (6 KB)
    __shared__ float smax[WPB];

    const int lane = threadIdx.x & 31;
    const int wave = threadIdx.x >> 5;
    const int ray  = blockIdx.x * WPB + wave;

    const float* rowD = density + (size_t)ray * S_SAMP;
    const float* rowZ = depth   + (size_t)ray * S_SAMP;
    const float* rowC = color   + (size_t)ray * S_SAMP * 3;

    // Issue the entire density stream asynchronously (6 in-flight ASYNC ops),
    // and prefetch the whole color stream into WGP-level caches.
#pragma unroll
    for (int c = 0; c < N_CHUNKS; ++c) {
        async_load_f32_to_lds(rowD + c * 32 + lane,
                              lds_off_u32(&sden[c][wave][lane]));
        __builtin_prefetch(rowC + (size_t)c * 96 + lane * 3, 0, 3);
    }

    float carry = 0.0f;                       // running sum of sigma*dt
    float c0 = 0.0f, c1 = 0.0f, c2 = 0.0f;    // RGB accumulators
    float dsum = 0.0f;                        // depth accumulator

#pragma unroll
    for (int c = 0; c < N_CHUNKS; ++c) {
        // Async loads complete in order: chunk c is resident once cnt <= 5-c.
        wait_async_le(N_CHUNKS - 1 - c);

        const int s     = c * 32 + lane;
        const float sig = sden[c][wave][lane];

        // delta_t: forward difference, 1e10 sentinel on the very last sample.
        float d  = rowZ[s];
        float dn = __shfl_down(d, 1, 32);
        if (lane == 31) dn = (c < N_CHUNKS - 1) ? rowZ[s + 1] : d + FARDELTA;
        const float sdt = sig * (dn - d);

        // Inclusive wave scan of sigma*dt (Hillis-Steele, 5 steps).
        float incl = sdt;
#pragma unroll
        for (int off = 1; off < 32; off <<= 1) {
            float t = __shfl_up(incl, off, 32);
            if (lane >= off) incl += t;
        }
        const float tot = incl + carry;       // inclusive prefix incl. prior chunks
        // w = T_excl * alpha = exp(-(tot - sdt)) - exp(-tot)
        const float w = __expf(sdt - tot) - __expf(-tot);
        carry = __shfl(tot, 31, 32);          // broadcast running total

        const F3 rgb = *reinterpret_cast<const F3*>(rowC + 3 * s);  // b96 load
        c0   += w * rgb.x;
        c1   += w * rgb.y;
        c2   += w * rgb.z;
        dsum += w * d;
    }

    // Wave reduction of the four accumulators.
#pragma unroll
    for (int off = 16; off >= 1; off >>= 1) {
        c0   += __shfl_down(c0,   off, 32);
        c1   += __shfl_down(c1,   off, 32);
        c2   += __shfl_down(c2,   off, 32);
        dsum += __shfl_down(dsum, off, 32);
    }

    if (lane == 0) {
        float* o = out + (size_t)ray * 4;
        o[0] = c0; o[1] = c1; o[2] = c2;
        o[3] = dsum;                          // unnormalized; pass2 divides by max
        smax[wave] = dsum;
    }
    __syncthreads();
    if (threadIdx.x == 0) {
        float m = smax[0];
#pragma unroll
        for (int i = 1; i < WPB; ++i) m = fmaxf(m, smax[i]);
        // dsum >= 0 always, so float order == unsigned-bits order.
        atomicMax(ws_max, __float_as_uint(m));
    }
}

__global__ __launch_bounds__(256) void vr_pass2(float* __restrict__ out,
                                                const unsigned* __restrict__ ws_max)
{
    const int r = blockIdx.x * blockDim.x + threadIdx.x;
    const float inv = 1.0f / __uint_as_float(*ws_max);
    out[(size_t)r * 4 + 3] *= inv;
}

__global__ void vr_init(unsigned* ws_max) { *ws_max = 0u; }

extern "C" void kernel_launch(void* const* d_in, const int* in_sizes, int n_in,
                              void* d_out, int out_size, void* d_ws, size_t ws_size,
                              hipStream_t stream) {
    const float* density = (const float*)d_in[0];   // [R, S]
    const float* color   = (const float*)d_in[1];   // [R, S, 3]
    const float* depth   = (const float*)d_in[2];   // [R, S]
    float*    out   = (float*)d_out;                // [R, 4]
    unsigned* wsmax = (unsigned*)d_ws;              // 4 bytes used

    vr_init<<<1, 1, 0, stream>>>(wsmax);
    vr_pass1<<<R_RAYS / WPB, 256, 0, stream>>>(density, color, depth, out, wsmax);
    vr_pass2<<<R_RAYS / 256, 256, 0, stream>>>(out, wsmax);
}